// Net_8340826489610
// MI455X (gfx1250) — compile-verified
//
#include <hip/hip_runtime.h>
#include <hip/hip_bf16.h>

// ---------------------------------------------------------------------------
// Net: GIN conv -> TopK(min_score) pool -> GIN conv -> gsum readout + KL loss
// N=50000 nodes, E=800000 edges, F=64, H=256, D=64, G=128 graphs
// fp32 throughout (softmax/log/KL need it) -> V_WMMA_F32_16X16X4_F32 for GEMMs
// ---------------------------------------------------------------------------

#define NN 50000
#define EE 800000
#define FF 64
#define HH 256
#define DD 64
#define GG 128

typedef __attribute__((ext_vector_type(2))) float v2f;
typedef __attribute__((ext_vector_type(8))) float v8f;
typedef int v4i __attribute__((vector_size(16)));

static __device__ __forceinline__ v8f wmma_f32(v2f a, v2f b, v8f c) {
  // D = A(16x4,f32) * B(4x16,f32) + C(16x16,f32)
  return __builtin_amdgcn_wmma_f32_16x16x4_f32(
      /*neg_a=*/false, a, /*neg_b=*/false, b,
      /*c_mod=*/(short)0, c, /*reuse_a=*/false, /*reuse_b=*/false);
}

// gfx1250 async memory->LDS path (guarded; falls back to float4 copy)
#if defined(__has_builtin)
#if __has_builtin(__builtin_amdgcn_global_load_async_to_lds_b128)
#define HAVE_ASYNC_LDS 1
#endif
#if __has_builtin(__builtin_amdgcn_s_wait_asynccnt)
#define HAVE_WAIT_ASYNC 1
#endif
#endif

static __device__ __forceinline__ void wait_asynccnt0() {
#if defined(HAVE_WAIT_ASYNC)
  __builtin_amdgcn_s_wait_asynccnt(0);
#else
  asm volatile("s_wait_asynccnt 0x0" ::: "memory");
#endif
}

// order-preserving float<->uint keys for atomicMax on floats
static __device__ __forceinline__ unsigned fkey(float f) {
  unsigned u = __float_as_uint(f);
  return (u & 0x80000000u) ? ~u : (u | 0x80000000u);
}
static __device__ __forceinline__ float funkey(unsigned u) {
  return (u & 0x80000000u) ? __uint_as_float(u & 0x7fffffffu)
                           : __uint_as_float(~u);
}

// ---------------------------------------------------------------------------
__global__ void zero_kernel(unsigned* __restrict__ p, int n) {
  int i = blockIdx.x * blockDim.x + threadIdx.x;
  if (i < n) p[i] = 0u;
}

// Repack a row-major weight W[K, nCols] into WMMA B-fragment-major order:
// out[(kstep*nTiles + tile)*32 + lane] = { W[k+koff, col], W[k+koff+1, col] }
// with k = 4*kstep, koff = (lane<16 ? 0 : 2), col = tile*16 + (lane&15).
// One coalesced b64 load per B fragment in the GEMM inner loop.
__global__ void wfrag_kernel(const float* __restrict__ W, v2f* __restrict__ out,
                             int nCols, int nKsteps, int nTiles) {
  int t = blockIdx.x * blockDim.x + threadIdx.x;
  int total = nKsteps * nTiles * 32;
  if (t >= total) return;
  int lane = t & 31;
  int tile = (t >> 5) % nTiles;
  int kstep = (t >> 5) / nTiles;
  int koff = (lane < 16) ? 0 : 2;
  int col = tile * 16 + (lane & 15);
  int row = kstep * 4 + koff;
  v2f v;
  v.x = W[row * nCols + col];
  v.y = W[(row + 1) * nCols + col];
  out[t] = v;
}

// s[n] = dot(x[n,:], pw); per-graph running max via ordered-uint atomicMax.
// One wave32 per node, 2 features per lane.
__global__ void score_kernel(const float* __restrict__ x,
                             const float* __restrict__ pw,
                             const int* __restrict__ batch,
                             float* __restrict__ sbuf,
                             unsigned* __restrict__ smaxu, int n) {
  int node = (blockIdx.x * blockDim.x + threadIdx.x) >> 5;
  int lane = threadIdx.x & 31;
  if (node >= n) return;
  float v = x[node * FF + lane] * pw[lane] +
            x[node * FF + 32 + lane] * pw[32 + lane];
  for (int o = 16; o > 0; o >>= 1) v += __shfl_down(v, o, 32);
  if (lane == 0) {
    sbuf[node] = v;
    atomicMax(&smaxu[batch[node]], fkey(v));
  }
}

// e[n] = exp(s - smax[batch]); per-graph sum
__global__ void exp_kernel(const float* __restrict__ sbuf,
                           const unsigned* __restrict__ smaxu,
                           const int* __restrict__ batch,
                           float* __restrict__ score,
                           float* __restrict__ esum, int n) {
  int i = blockIdx.x * blockDim.x + threadIdx.x;
  if (i >= n) return;
  int g = batch[i];
  float e = expf(sbuf[i] - funkey(smaxu[g]));
  score[i] = e;
  atomicAdd(&esum[g], e);
}

// score = e/sum; thresh = min(1/sum - TOL, MIN_SCORE) (max e is exactly 1.0);
// mask, KL accumulation, kept-node counts
__global__ void mask_kernel(const float* __restrict__ esum,
                            const int* __restrict__ batch,
                            const float* __restrict__ na,
                            float* __restrict__ score,
                            float* __restrict__ mfb,
                            float* __restrict__ klsum,
                            float* __restrict__ cnt,
                            float* __restrict__ ctot, int n) {
  int i = blockIdx.x * blockDim.x + threadIdx.x;
  if (i >= n) return;
  int g = batch[i];
  float sum = esum[g];
  float sc = score[i] / sum;
  float thresh = fminf(1.0f / sum - 1e-7f, 0.05f);
  float m = (sc > thresh) ? 1.0f : 0.0f;
  score[i] = sc;
  mfb[i] = m;
  if (m != 0.0f) {
    float t = na[i];
    float tlogt = (t > 0.0f) ? t * logf(t) : 0.0f;
    float kl = tlogt - t * logf(sc + 1e-14f);
    atomicAdd(&klsum[g], kl);
    atomicAdd(&cnt[g], 1.0f);
    atomicAdd(ctot, 1.0f);
  }
}

// vectorized copy (z1 = x)
__global__ void copy4_kernel(const float* __restrict__ in,
                             float* __restrict__ out, int n4) {
  int t = blockIdx.x * blockDim.x + threadIdx.x;
  if (t < n4) ((float4*)out)[t] = ((const float4*)in)[t];
}

// Z[dst] += X[src] over edges (optionally gated by node mask).
// 16 threads per edge, float4 per thread -> 51.2M L2-resident f32 atomics.
__global__ void edge_agg_kernel(const float* __restrict__ X,
                                const int* __restrict__ ei,
                                float* __restrict__ Zo,
                                const float* __restrict__ mf, int e_cnt) {
  int t = blockIdx.x * blockDim.x + threadIdx.x;
  int e = t >> 4;
  if (e >= e_cnt) return;
  int s = ei[e];
  int d = ei[e_cnt + e];
  if (mf != nullptr) {
    if (mf[s] == 0.0f || mf[d] == 0.0f) return;
  }
  int c = (t & 15) * 4;
  float4 v = *(const float4*)(X + (size_t)s * FF + c);
  float* o = Zo + (size_t)d * FF + c;
  atomicAdd(o + 0, v.x);
  atomicAdd(o + 1, v.y);
  atomicAdd(o + 2, v.z);
  atomicAdd(o + 3, v.w);
}

// ---------------------------------------------------------------------------
// Fused 2-layer MLP on 16-row node tiles with V_WMMA_F32_16X16X4_F32:
//   mid = relu(Z[16,64] @ Wa[64,256] + ba)   (staged in LDS)
//   out = relu(mid @ Wb[256,64] + bb)
// Weights come pre-packed in B-fragment-major order (one b64 load per frag).
// mode 0: store out[row,col]
// mode 1: out *= mf[row]; atomicAdd into xgsum[batch[row]*64+col] (readout)
// Block = 256 threads = 8 waves. Waves each own 16-col output tiles.
// LDS padded (LDA=68, LDM=260) so 16-lane column reads hit distinct banks.
// ---------------------------------------------------------------------------
#define LDA 68
#define LDM 260

__global__ __launch_bounds__(256) void mlp_kernel(
    const float* __restrict__ Z,
    const v2f* __restrict__ Waf,      // [FF/4][16][32] frags of Wa
    const float* __restrict__ ba,
    const v2f* __restrict__ Wbf,      // [HH/4][4][32] frags of Wb
    const float* __restrict__ bb,
    float* __restrict__ Out,
    const float* __restrict__ mf,
    const int* __restrict__ batch,
    float* __restrict__ xgsum, int numTiles, int mode) {
  __shared__ float As[16 * LDA];   // 16x64 input tile
  __shared__ float Am[16 * LDM];   // 16x256 mid activations
  const int tid = threadIdx.x;
  const int lane = tid & 31;
  const int w = tid >> 5;
  const int mrowA = lane & 15;              // A-matrix row for this lane
  const int koff = (lane < 16) ? 0 : 2;     // K pair offset (ISA A/B layout)
  const int rbase = (lane < 16) ? 0 : 8;    // C/D row base (ISA C layout)

  for (int tile = blockIdx.x; tile < numTiles; tile += gridDim.x) {
    const int rowBase = tile * 16;
    // phase 1: stage the 16x64 A tile into LDS (async path on gfx1250)
    {
      int idx = tid * 4;
      int r = idx >> 6, c = idx & 63;
      const float* gsrc = Z + (size_t)(rowBase + r) * FF + c;
      float* ldst = As + r * LDA + c;
#if defined(HAVE_ASYNC_LDS)
      __builtin_amdgcn_global_load_async_to_lds_b128(
          (__attribute__((address_space(1))) v4i*)gsrc,
          (__attribute__((address_space(3))) v4i*)ldst, 0, 0);
      wait_asynccnt0();
#else
      *(float4*)ldst = *(const float4*)gsrc;
#endif
    }
    __syncthreads();

    // phase 2: mid = relu(A @ Wa + ba); each wave does col tiles w and w+8
    #pragma unroll
    for (int t = 0; t < 2; ++t) {
      const int tileIdx = w + t * 8;
      const int col = tileIdx * 16 + (lane & 15);
      v8f acc = {};
      #pragma unroll 4
      for (int k = 0; k < FF; k += 4) {
        v2f a;
        a.x = As[mrowA * LDA + k + koff];
        a.y = As[mrowA * LDA + k + koff + 1];
        v2f b = Waf[(((k >> 2) * 16) + tileIdx) * 32 + lane];
        acc = wmma_f32(a, b, acc);
      }
      const float bias = ba[col];
      #pragma unroll
      for (int r = 0; r < 8; ++r)
        Am[(rbase + r) * LDM + col] = fmaxf(acc[r] + bias, 0.0f);
    }
    __syncthreads();

    // phase 3: out = relu(mid @ Wb + bb); waves 0..3 own the 4 col tiles of D
    if (w < 4) {
      const int col = w * 16 + (lane & 15);
      v8f acc = {};
      #pragma unroll 4
      for (int k = 0; k < HH; k += 4) {
        v2f a;
        a.x = Am[mrowA * LDM + k + koff];
        a.y = Am[mrowA * LDM + k + koff + 1];
        v2f b = Wbf[(((k >> 2) * 4) + w) * 32 + lane];
        acc = wmma_f32(a, b, acc);
      }
      const float bias = bb[col];
      #pragma unroll
      for (int r = 0; r < 8; ++r) {
        const int row = rowBase + rbase + r;
        float val = fmaxf(acc[r] + bias, 0.0f);
        if (mode == 0) {
          Out[(size_t)row * DD + col] = val;
        } else {
          val *= mf[row];
          atomicAdd(&xgsum[batch[row] * DD + col], val);
        }
      }
    }
    __syncthreads();
  }
}

// xp = h1 * score * mf ; z2 initialized to xp (in-place safe over h1)
__global__ void pool_kernel(const float* __restrict__ h1,
                            const float* __restrict__ score,
                            const float* __restrict__ mf,
                            float* __restrict__ xp,
                            float* __restrict__ z2, int total) {
  int t = blockIdx.x * blockDim.x + threadIdx.x;
  if (t >= total) return;
  int node = t >> 6;
  float v = h1[t] * (score[node] * mf[node]);
  xp[t] = v;
  z2[t] = v;
}

// out[g]      = dot(xgsum[g,:], Wl) + bl
// out[128+g]  = klsum[g] / max(cnt[g],1)
// out[256]    = kept/N
__global__ void finalize_kernel(const float* __restrict__ xgsum,
                                const float* __restrict__ Wl,
                                const float* __restrict__ bl,
                                const float* __restrict__ klsum,
                                const float* __restrict__ cnt,
                                const float* __restrict__ ctot,
                                float* __restrict__ out) {
  int g = (blockIdx.x * blockDim.x + threadIdx.x) >> 5;
  int lane = threadIdx.x & 31;
  if (g >= GG) return;
  float v = xgsum[g * DD + lane] * Wl[lane] +
            xgsum[g * DD + 32 + lane] * Wl[32 + lane];
  for (int o = 16; o > 0; o >>= 1) v += __shfl_down(v, o, 32);
  if (lane == 0) {
    out[g] = v + bl[0];
    out[GG + g] = klsum[g] / fmaxf(cnt[g], 1.0f);
    if (g == 0) out[2 * GG] = ctot[0] * (1.0f / (float)NN);
  }
}

// ---------------------------------------------------------------------------
extern "C" void kernel_launch(void* const* d_in, const int* in_sizes, int n_in,
                              void* d_out, int out_size, void* d_ws,
                              size_t ws_size, hipStream_t stream) {
  const float* x   = (const float*)d_in[0];   // [N,64]
  const float* na  = (const float*)d_in[1];   // [N]
  const float* W1a = (const float*)d_in[2];   // [64,256]
  const float* b1a = (const float*)d_in[3];   // [256]
  const float* W1b = (const float*)d_in[4];   // [256,64]
  const float* b1b = (const float*)d_in[5];   // [64]
  const float* pw  = (const float*)d_in[6];   // [64]
  const float* W2a = (const float*)d_in[7];   // [64,256]
  const float* b2a = (const float*)d_in[8];   // [256]
  const float* W2b = (const float*)d_in[9];   // [256,64]
  const float* b2b = (const float*)d_in[10];  // [64]
  const float* Wl  = (const float*)d_in[11];  // [64]
  const float* bl  = (const float*)d_in[12];  // [1]
  const int* ei    = (const int*)d_in[13];    // [2,E] (src row, dst row)
  const int* batch = (const int*)d_in[14];    // [N]
  float* out = (float*)d_out;                 // 128 + 128 + 1

  // workspace layout (floats)
  float* ws = (float*)d_ws;
  float* score = ws;                         // N  (e, then softmax score)
  float* mfb   = score + NN;                 // N  mask as float
  float* sbuf  = mfb + NN;                   // N  raw attention logits
  float* bufA  = sbuf + NN;                  // N*64  (z1, later xp)
  float* bufB  = bufA + (size_t)NN * FF;     // N*64  (h1, later z2)
  v2f* w1af    = (v2f*)(bufB + (size_t)NN * FF); // 16*16*32 frags
  v2f* w1bf    = w1af + 16 * 16 * 32;            // 64*4*32 frags
  v2f* w2af    = w1bf + 64 * 4 * 32;
  v2f* w2bf    = w2af + 16 * 16 * 32;
  float* xgsum = (float*)(w2bf + 64 * 4 * 32);   // G*64 readout accumulators
  float* klsum = xgsum + GG * DD;            // G
  float* cnt   = klsum + GG;                 // G
  float* esum  = cnt + GG;                   // G
  float* ctot  = esum + GG;                  // 1
  unsigned* smaxu = (unsigned*)(ctot + 1);   // G (ordered-uint max keys)

  // 0) zero all accumulators (xgsum .. smaxu are contiguous)
  const int accWords = GG * DD + 3 * GG + 1 + GG;
  zero_kernel<<<(accWords + 255) / 256, 256, 0, stream>>>((unsigned*)xgsum,
                                                          accWords);
  // 0b) pack weights into B-fragment-major order
  wfrag_kernel<<<32, 256, 0, stream>>>(W1a, w1af, HH, FF / 4, HH / 16);
  wfrag_kernel<<<32, 256, 0, stream>>>(W1b, w1bf, DD, HH / 4, DD / 16);
  wfrag_kernel<<<32, 256, 0, stream>>>(W2a, w2af, HH, FF / 4, HH / 16);
  wfrag_kernel<<<32, 256, 0, stream>>>(W2b, w2bf, DD, HH / 4, DD / 16);
  // 1) s = x @ pw, per-graph max
  score_kernel<<<(NN * 32 + 255) / 256, 256, 0, stream>>>(x, pw, batch, sbuf,
                                                          smaxu, NN);
  // 2) e = exp(s - smax), per-graph sum
  exp_kernel<<<(NN + 255) / 256, 256, 0, stream>>>(sbuf, smaxu, batch, score,
                                                   esum, NN);
  // 3) softmax score, mask, KL accumulation
  mask_kernel<<<(NN + 255) / 256, 256, 0, stream>>>(esum, batch, na, score,
                                                    mfb, klsum, cnt, ctot, NN);
  // 4) z1 = x
  copy4_kernel<<<(NN * FF / 4 + 255) / 256, 256, 0, stream>>>(x, bufA,
                                                              NN * FF / 4);
  // 5) z1 += scatter-add of x over edges
  edge_agg_kernel<<<(EE * 16 + 255) / 256, 256, 0, stream>>>(x, ei, bufA,
                                                             nullptr, EE);
  // 6) h1 = relu(relu(z1@W1a+b1a)@W1b+b1b)  [WMMA f32]
  mlp_kernel<<<1024, 256, 0, stream>>>(bufA, w1af, b1a, w1bf, b1b, bufB,
                                       nullptr, nullptr, nullptr, NN / 16, 0);
  // 7) xp = h1 * score * mask ; z2 = xp
  pool_kernel<<<(NN * FF + 255) / 256, 256, 0, stream>>>(bufB, score, mfb,
                                                         bufA, bufB, NN * FF);
  // 8) z2 += masked scatter-add of xp over surviving edges
  edge_agg_kernel<<<(EE * 16 + 255) / 256, 256, 0, stream>>>(bufA, ei, bufB,
                                                             mfb, EE);
  // 9) h2 MLP [WMMA f32], fused *mask and graph-sum readout
  mlp_kernel<<<1024, 256, 0, stream>>>(bufB, w2af, b2a, w2bf, b2b, nullptr,
                                       mfb, batch, xgsum, NN / 16, 1);
  // 10) head + KL mean + ratio
  finalize_kernel<<<(GG * 32 + 255) / 256, 256, 0, stream>>>(
      xgsum, Wl, bl, klsum, cnt, ctot, out);

  (void)in_sizes; (void)n_in; (void)out_size; (void)ws_size;
}